// KnnBruteforce_2568390443357
// MI455X (gfx1250) — compile-verified
//
#include <hip/hip_runtime.h>

typedef __attribute__((ext_vector_type(2))) float v2f;
typedef __attribute__((ext_vector_type(4))) float v4f;
typedef __attribute__((ext_vector_type(8))) float v8f;

#define BATCH 4
#define NPTS  8192
#define KNN   16
#define WAVES_PER_BLOCK 8
#define QUERIES_PER_WAVE 32
#define QUERIES_PER_BLOCK (WAVES_PER_BLOCK * QUERIES_PER_WAVE)   // 256

__global__ __launch_bounds__(256) void knn_bruteforce_wmma(
    const float* __restrict__ pos,      // [B, 3, N]
    int*         __restrict__ nn_out,   // [B, K, N]
    float*       __restrict__ dist_out) // [B, K, N]
{
    // Per-wave LDS regions (no cross-wave sharing -> no block barriers needed)
    __shared__ float sDist[WAVES_PER_BLOCK][2][256];      // 16 KB: d2 tiles
    __shared__ float sTopD[WAVES_PER_BLOCK][32][KNN];     // 16 KB: top-k dists
    __shared__ int   sTopI[WAVES_PER_BLOCK][32][KNN];     // 16 KB: top-k idx

    const int lane  = threadIdx.x & 31;
    const int wave  = threadIdx.x >> 5;
    const int b     = blockIdx.y;
    const int qbase = blockIdx.x * QUERIES_PER_BLOCK + wave * QUERIES_PER_WAVE;

    const float* X = pos + (size_t)b * 3 * NPTS;
    const float* Y = X + NPTS;
    const float* Z = Y + NPTS;

    const int half = lane >> 4;     // 0: K0/K1 lanes, 1: K2/K3 lanes
    const int l15  = lane & 15;

    // ---- A fragments (fixed per wave): two 16-row query tiles ----
    // 16x4 f32 A layout: VGPR0 = {K0 | K2}, VGPR1 = {K1 | K3}; M = lane%16
    v2f a0, a1;
    {
        const int m0 = qbase + l15;
        const int m1 = qbase + 16 + l15;
        a0.x = half ? Z[m0] : X[m0];
        a0.y = half ? 0.0f  : Y[m0];
        a1.x = half ? Z[m1] : X[m1];
        a1.y = half ? 0.0f  : Y[m1];
    }

    // Per-lane squared norms of the 8 query rows this lane's C-VGPRs hold,
    // for both row tiles: row m = qbase + 16*t + r + 8*half
    float sqA[2][8];
#pragma unroll
    for (int t = 0; t < 2; ++t) {
#pragma unroll
        for (int r = 0; r < 8; ++r) {
            const int m = qbase + 16 * t + r + 8 * half;
            const float x = X[m], y = Y[m], z = Z[m];
            sqA[t][r] = x * x + y * y + z * z;
        }
    }

    // ---- init top-k list (per-lane list lives in LDS) ----
#pragma unroll
    for (int i = 0; i < KNN; ++i) {
        sTopD[wave][lane][i] = __builtin_huge_valf();
        sTopI[wave][lane][i] = 0;
    }
    float curMax    = __builtin_huge_valf();
    int   curMaxPos = 0;

    // ---- sweep all candidates in 16-wide column tiles ----
    for (int j0 = 0; j0 < NPTS; j0 += 16) {
        const int j = j0 + l15;
        const float xj = X[j], yj = Y[j], zj = Z[j];
        const float sqJ = xj * xj + yj * yj + zj * zj;

        v2f bf;
        bf.x = half ? zj   : xj;
        bf.y = half ? 0.0f : yj;

        v8f c0 = {};
        v8f c1 = {};
        // D = A x B (+0): 16x16 tile of dot products, f32 exact
        c0 = __builtin_amdgcn_wmma_f32_16x16x4_f32(
                false, a0, false, bf, (short)0, c0, false, false);
        c1 = __builtin_amdgcn_wmma_f32_16x16x4_f32(
                false, a1, false, bf, (short)0, c1, false, false);

        // d2 = |xi|^2 + |xj|^2 - 2*dot ; scatter tile rows into LDS
#pragma unroll
        for (int r = 0; r < 8; ++r) {
            const int row = r + 8 * half;
            sDist[wave][0][row * 16 + l15] = sqA[0][r] + sqJ - 2.0f * c0[r];
            sDist[wave][1][row * 16 + l15] = sqA[1][r] + sqJ - 2.0f * c1[r];
        }

        // single wave: lockstep issue + in-order LDS => dscnt wait suffices
        asm volatile("s_wait_dscnt 0" ::: "memory");

        // lane owns query (local) = lane: tile = half, in-tile row = l15.
        // Bulk-load the 16-value row with 4x ds_load_b128 into registers.
        const v4f* myRow4 = (const v4f*)&sDist[wave][half][l15 * 16];
        const v4f r0 = myRow4[0];
        const v4f r1 = myRow4[1];
        const v4f r2 = myRow4[2];
        const v4f r3 = myRow4[3];

        float d[16];
#pragma unroll
        for (int c = 0; c < 4; ++c) {
            d[c]      = r0[c];
            d[4 + c]  = r1[c];
            d[8 + c]  = r2[c];
            d[12 + c] = r3[c];
        }

        // tile minimum via min tree (registers only)
        v4f m01 = __builtin_elementwise_min(r0, r1);
        v4f m23 = __builtin_elementwise_min(r2, r3);
        v4f m   = __builtin_elementwise_min(m01, m23);
        float tileMin = __builtin_fminf(__builtin_fminf(m.x, m.y),
                                        __builtin_fminf(m.z, m.w));

        // fast path: nothing in this tile beats the current k-th distance
        if (tileMin < curMax) {
#pragma unroll
            for (int c = 0; c < 16; ++c) {
                if (d[c] < curMax) {
                    sTopD[wave][lane][curMaxPos] = d[c];
                    sTopI[wave][lane][curMaxPos] = j0 + c;
                    // rescan list for the new max
                    float mv = sTopD[wave][lane][0];
                    int   mp = 0;
#pragma unroll
                    for (int i2 = 1; i2 < KNN; ++i2) {
                        const float v = sTopD[wave][lane][i2];
                        if (v > mv) { mv = v; mp = i2; }
                    }
                    curMax = mv;
                    curMaxPos = mp;
                }
            }
        }
        // next iteration's LDS stores are same-wave in-order vs these loads
    }

    // ---- sort the 16 survivors ascending (matches top_k output order) ----
    for (int a = 0; a < KNN; ++a) {
        float mv = sTopD[wave][lane][a];
        int   mp = a;
        for (int c = a + 1; c < KNN; ++c) {
            const float v = sTopD[wave][lane][c];
            if (v < mv) { mv = v; mp = c; }
        }
        const float td = sTopD[wave][lane][a];
        const int   ti = sTopI[wave][lane][a];
        sTopD[wave][lane][a]  = sTopD[wave][lane][mp];
        sTopI[wave][lane][a]  = sTopI[wave][lane][mp];
        sTopD[wave][lane][mp] = td;
        sTopI[wave][lane][mp] = ti;
    }

    // ---- coalesced writes, output layout [B, K, N] ----
    const int q = qbase + lane;
    const size_t outBase = (size_t)b * KNN * NPTS;
#pragma unroll
    for (int kk = 0; kk < KNN; ++kk) {
        nn_out  [outBase + (size_t)kk * NPTS + q] = sTopI[wave][lane][kk];
        dist_out[outBase + (size_t)kk * NPTS + q] = sTopD[wave][lane][kk];
    }
}

extern "C" void kernel_launch(void* const* d_in, const int* in_sizes, int n_in,
                              void* d_out, int out_size, void* d_ws, size_t ws_size,
                              hipStream_t stream) {
    const float* pos = (const float*)d_in[0];
    // tuple output: NN (int32) then distances (f32), each B*K*N flat
    int*   nn   = (int*)d_out;
    float* dist = (float*)d_out + (size_t)BATCH * KNN * NPTS;

    dim3 grid(NPTS / QUERIES_PER_BLOCK, BATCH);
    knn_bruteforce_wmma<<<grid, 256, 0, stream>>>(pos, nn, dist);
}